// GResidule_52407190946013
// MI455X (gfx1250) — compile-verified
//
#include <hip/hip_runtime.h>
#include <math.h>

typedef __attribute__((ext_vector_type(2))) float v2f;
typedef __attribute__((ext_vector_type(8))) float v8f;

#define HEADS 4
#define DCH   128
#define CCH   32
#define NEG_SLOPE 0.2f

// ---------------------------------------------------------------------------
// h = X[N,128] @ W[128,128] via V_WMMA_F32_16X16X4_F32.
// One wave per 16x16 output tile; 8 waves (256 thr) per block cover one 16-row
// strip across all 128 output columns. K looped in steps of 4.
// A layout (16x4 f32): lanes 0-15 M=0..15 hold K=k0,k0+1; lanes 16-31 hold K=k0+2,k0+3.
// B layout mirrored over N. D layout: VGPR r = row (r + 8*(lane>>4)), col = lane&15.
// ---------------------------------------------------------------------------
__global__ void gemm128_wmma_kernel(const float* __restrict__ X,
                                    const float* __restrict__ W,
                                    float* __restrict__ Hout, int N) {
  const int lane  = threadIdx.x & 31;
  const int wave  = threadIdx.x >> 5;          // 0..7 -> 16-col tile
  const int mBase = blockIdx.x * 16;
  const int nBase = wave * 16;
  const int half  = lane >> 4;                  // 0 or 1
  const int l     = lane & 15;

  int m = mBase + l;
  if (m >= N) m = N - 1;                        // clamp loads; EXEC stays all-1s
  const float* xrow = X + (size_t)m * DCH;
  const int n = nBase + l;

  v8f acc = {0.f, 0.f, 0.f, 0.f, 0.f, 0.f, 0.f, 0.f};
#pragma unroll
  for (int k0 = 0; k0 < DCH; k0 += 4) {
    const int ka = k0 + 2 * half;
    v2f a, b;
    a.x = xrow[ka];
    a.y = xrow[ka + 1];
    b.x = W[(size_t)ka * DCH + n];
    b.y = W[(size_t)(ka + 1) * DCH + n];
    acc = __builtin_amdgcn_wmma_f32_16x16x4_f32(
        /*neg_a=*/false, a, /*neg_b=*/false, b,
        /*c_mod=*/(short)0, acc, /*reuse_a=*/false, /*reuse_b=*/false);
  }
#pragma unroll
  for (int r = 0; r < 8; ++r) {
    const int mr = mBase + r + 8 * half;
    if (mr < N) Hout[(size_t)mr * DCH + nBase + l] = acc[r];
  }
}

// ---------------------------------------------------------------------------
// alpha_src[n,h] = sum_c h[n,h,c]*a_src[h,c]; same for a_dst.
// One wave per (n,h); lane = channel; wave32 shfl_xor reduction.
// ---------------------------------------------------------------------------
__global__ void alpha_kernel(const float* __restrict__ Hm,
                             const float* __restrict__ a_src,
                             const float* __restrict__ a_dst,
                             float* __restrict__ as, float* __restrict__ ad,
                             int N) {
  const int gwave = (int)((blockIdx.x * (size_t)blockDim.x + threadIdx.x) >> 5);
  const int lane  = threadIdx.x & 31;
  if (gwave >= N * HEADS) return;               // wave-uniform exit
  const int node = gwave >> 2;
  const int hh   = gwave & 3;
  const float hv = Hm[(size_t)node * DCH + hh * CCH + lane];
  float s = hv * a_src[hh * CCH + lane];
  float d = hv * a_dst[hh * CCH + lane];
#pragma unroll
  for (int off = 16; off > 0; off >>= 1) {
    s += __shfl_xor(s, off, 32);
    d += __shfl_xor(d, off, 32);
  }
  if (lane == 0) { as[gwave] = s; ad[gwave] = d; }
}

// Monotone float->uint key so atomicMax(u32) implements float max.
__device__ __forceinline__ unsigned f32_key(float f) {
  unsigned u = __float_as_uint(f);
  return (u & 0x80000000u) ? ~u : (u ^ 0x80000000u);
}

// ---------------------------------------------------------------------------
// Per (edge, head): logits = leaky_relu(as[src]+ad[dst]); write rel output;
// atomicMax the key into mkey[dst,h].
// ---------------------------------------------------------------------------
__global__ void logits_kernel(const int* __restrict__ src,
                              const int* __restrict__ dst,
                              const float* __restrict__ as,
                              const float* __restrict__ ad,
                              float* __restrict__ rel_out,
                              unsigned* __restrict__ mkey, int E) {
  const long t = blockIdx.x * (long)blockDim.x + threadIdx.x;
  if (t >= (long)E * HEADS) return;
  const int e  = (int)(t >> 2);
  const int hh = (int)(t & 3);
  const int s  = src[e];
  const int d  = dst[e];
  const float x  = as[s * HEADS + hh] + ad[d * HEADS + hh];
  const float lg = x > 0.f ? x : NEG_SLOPE * x;
  rel_out[t] = lg;
  atomicMax(&mkey[d * HEADS + hh], f32_key(lg));
}

// Decode per-(node,head) max key; untouched (0) or non-finite -> 0.
__global__ void decode_m_kernel(const unsigned* __restrict__ mkey,
                                float* __restrict__ mval, int NH) {
  const int t = blockIdx.x * blockDim.x + threadIdx.x;
  if (t >= NH) return;
  const unsigned key = mkey[t];
  float m = 0.f;
  if (key != 0u) {
    const unsigned u = (key & 0x80000000u) ? (key ^ 0x80000000u) : ~key;
    m = __uint_as_float(u);
    if (!isfinite(m)) m = 0.f;
  }
  mval[t] = m;
}

// e = exp(logit - m[dst]); stash e in attn output slot; atomicAdd into denom.
__global__ void exp_kernel(const int* __restrict__ dst,
                           const float* __restrict__ rel,
                           const float* __restrict__ mval,
                           float* __restrict__ e_out,
                           float* __restrict__ denom, int E) {
  const long t = blockIdx.x * (long)blockDim.x + threadIdx.x;
  if (t >= (long)E * HEADS) return;
  const int e  = (int)(t >> 2);
  const int hh = (int)(t & 3);
  const int d  = dst[e];
  const float ev = expf(rel[t] - mval[d * HEADS + hh]);
  e_out[t] = ev;
  atomicAdd(&denom[d * HEADS + hh], ev);
}

// ---------------------------------------------------------------------------
// One wave per (edge, head), lane = channel:
//   attn = e/(denom[dst]+1e-16); write attn; out[dst,h,c] += h[src,h,c]*attn
// Coalesced 128B gather of h[src]; coalesced atomic adds.
// ---------------------------------------------------------------------------
__global__ void scatter_kernel(const int* __restrict__ src,
                               const int* __restrict__ dst,
                               const float* __restrict__ Hm,
                               const float* __restrict__ denom,
                               float* __restrict__ attn_io,
                               float* __restrict__ out_acc, int E) {
  const long gwave = (blockIdx.x * (long)blockDim.x + threadIdx.x) >> 5;
  const int lane = threadIdx.x & 31;
  if (gwave >= (long)E * HEADS) return;         // wave-uniform exit
  const int e  = (int)(gwave >> 2);
  const int hh = (int)(gwave & 3);
  const int s  = src[e];
  const int d  = dst[e];
  const float ev = attn_io[gwave];              // broadcast load
  const float a  = ev / (denom[d * HEADS + hh] + 1e-16f);
  if (lane == 0) attn_io[gwave] = a;
  const float msg = Hm[(size_t)s * DCH + hh * CCH + lane] * a;
  atomicAdd(&out_acc[(size_t)d * DCH + hh * CCH + lane], msg);
}

__global__ void residual_kernel(const float* __restrict__ xin,
                                const float* __restrict__ out_acc,
                                float* __restrict__ xout, long n) {
  const long t = blockIdx.x * (long)blockDim.x + threadIdx.x;
  if (t >= n) return;
  xout[t] = xin[t] + out_acc[t];
}

__global__ void zero_kernel(float* __restrict__ p, long n) {
  const long t = blockIdx.x * (long)blockDim.x + threadIdx.x;
  if (t < n) p[t] = 0.f;
}

static inline unsigned grid1d(long n, int block) {
  return (unsigned)((n + block - 1) / block);
}

extern "C" void kernel_launch(void* const* d_in, const int* in_sizes, int n_in,
                              void* d_out, int out_size, void* d_ws, size_t ws_size,
                              hipStream_t stream) {
  const float* x0    = (const float*)d_in[0];
  const int*   ei    = (const int*)d_in[1];
  const float* W     = (const float*)d_in[2];
  const float* a_src = (const float*)d_in[3];
  const float* a_dst = (const float*)d_in[4];
  const int STEPS = 3;                          // reference constant

  const int N = in_sizes[0] / DCH;
  const int E = in_sizes[1] / 2;
  const int* src = ei;
  const int* dst = ei + E;

  float* out  = (float*)d_out;
  float* xs   = out;                                   // [STEPS, N, 128]
  float* attn = xs + (size_t)STEPS * N * DCH;          // [STEPS, E, 4]
  float* rel  = attn + (size_t)STEPS * E * HEADS;      // [STEPS, E, 4]

  // Workspace layout (floats): h | out_acc | denom | mkey | mval | as | ad
  float*    Hm      = (float*)d_ws;
  float*    out_acc = Hm + (size_t)N * DCH;
  float*    denom   = out_acc + (size_t)N * DCH;
  unsigned* mkey    = (unsigned*)(denom + (size_t)N * HEADS);
  float*    mval    = (float*)(mkey + (size_t)N * HEADS);
  float*    as      = mval + (size_t)N * HEADS;
  float*    ad      = as + (size_t)N * HEADS;

  const int BLK = 256;
  const float* xin = x0;

  for (int t = 0; t < STEPS; ++t) {
    // zero out_acc + denom + mkey (contiguous region)
    const long zn = (long)N * DCH + 2L * N * HEADS;
    zero_kernel<<<grid1d(zn, BLK), BLK, 0, stream>>>(out_acc, zn);

    // h = x @ W  (WMMA fp32)
    gemm128_wmma_kernel<<<grid1d((long)N, 16), BLK, 0, stream>>>(xin, W, Hm, N);

    // per-(node,head) head projections
    alpha_kernel<<<grid1d((long)N * HEADS * 32, BLK), BLK, 0, stream>>>(
        Hm, a_src, a_dst, as, ad, N);

    float* rel_t  = rel  + (size_t)t * E * HEADS;
    float* attn_t = attn + (size_t)t * E * HEADS;

    logits_kernel<<<grid1d((long)E * HEADS, BLK), BLK, 0, stream>>>(
        src, dst, as, ad, rel_t, mkey, E);

    decode_m_kernel<<<grid1d((long)N * HEADS, BLK), BLK, 0, stream>>>(
        mkey, mval, N * HEADS);

    exp_kernel<<<grid1d((long)E * HEADS, BLK), BLK, 0, stream>>>(
        dst, rel_t, mval, attn_t, denom, E);

    scatter_kernel<<<grid1d((long)E * HEADS * 32, BLK), BLK, 0, stream>>>(
        src, dst, Hm, denom, attn_t, out_acc, E);

    float* xs_t = xs + (size_t)t * N * DCH;
    residual_kernel<<<grid1d((long)N * DCH, BLK), BLK, 0, stream>>>(
        xin, out_acc, xs_t, (long)N * DCH);

    xin = xs_t;
  }
}